// MultiPeakAwareAttention_39350490366464
// MI455X (gfx1250) — compile-verified
//
#include <hip/hip_runtime.h>
#include <math.h>

typedef __attribute__((ext_vector_type(2))) float v2f;
typedef __attribute__((ext_vector_type(8))) float v8f;

namespace {
constexpr int BATCH = 2;
constexpr int HDIM = 64, WDIM = 64;
constexpr int N = 4096;          // H*W
constexpr int C = 128;
constexpr int TOPK = 8;
constexpr int SINK_ITERS = 5;
constexpr int RADIUS = 3;
constexpr int RCHUNK = 16;       // row chunks for column-LSE parallelism

// d_out layout (floats), reference return order:
constexpr size_t O_WARP = 0;                                 // [B,N,2]
constexpr size_t O_ENT  = O_WARP + (size_t)BATCH * N * 2;    // [B,N]
constexpr size_t O_SIM  = O_ENT  + (size_t)BATCH * N;        // [B,N,N]
constexpr size_t O_GEO  = O_SIM  + (size_t)BATCH * N * N;    // [B,N,K]

// workspace layout (float slots)
constexpr size_t W_NA = 0;                                   // normA [B,N,C]
constexpr size_t W_NB = W_NA + (size_t)BATCH * N * C;        // normB [B,N,C]
constexpr size_t W_U  = W_NB + (size_t)BATCH * N * C;        // u [B,N+1]
constexpr size_t W_V  = W_U  + (size_t)BATCH * (N + 1) + 64; // v [B,N+1]
constexpr size_t W_TV = W_V  + (size_t)BATCH * (N + 1) + 64; // topk vals [B,N,K]
constexpr size_t W_TI = W_TV + (size_t)BATCH * N * TOPK;     // topk idx (int) [B,N,K]
constexpr size_t W_TP = W_TI + (size_t)BATCH * N * TOPK;     // topk pos [B,N,K,2]
constexpr size_t W_PM = W_TP + (size_t)BATCH * N * TOPK * 2; // partial max [B,RCHUNK,N]
constexpr size_t W_PS = W_PM + (size_t)BATCH * RCHUNK * N;   // partial sum [B,RCHUNK,N]
} // namespace

// ---------------- row L2-normalization ----------------
__global__ __launch_bounds__(128) void k_rownorm(const float* __restrict__ in,
                                                 float* __restrict__ out) {
  int row = blockIdx.x;            // B*N rows
  int t = threadIdx.x;             // 128 threads = C
  float x = in[(size_t)row * C + t];
  float ss = x * x;
  #pragma unroll
  for (int o = 16; o > 0; o >>= 1) ss += __shfl_xor(ss, o, 32);
  __shared__ float part[4];
  int wave = t >> 5, lane = t & 31;
  if (lane == 0) part[wave] = ss;
  __syncthreads();
  float tot = part[0] + part[1] + part[2] + part[3];
  out[(size_t)row * C + t] = x / sqrtf(tot);
}

// ---------------- fp32 WMMA GEMM: sim = normA * normB^T ----------------
// Block = 256 threads (8 waves) computes a 64x64 C tile; A/B staged in LDS
// (K chunked at 64, padded 68-float row stride -> conflict-free fragment reads).
// Each wave owns two horizontally adjacent 16x16 tiles sharing the A fragment.
__global__ __launch_bounds__(256) void k_gemm_wmma(const float* __restrict__ A,
                                                   const float* __restrict__ B,
                                                   float* __restrict__ sim) {
  __shared__ float As[64 * 68];
  __shared__ float Bs[64 * 68];
  int blk = blockIdx.x;
  int b = blk >> 12;               // 4096 blocks per batch image
  int rem = blk & 4095;
  int bi = rem >> 6, bj = rem & 63;
  int t = threadIdx.x;
  int wave = t >> 5, lane = t & 31;
  int half = lane >> 4, lr = lane & 15;

  int ti = wave >> 1;              // A-tile row (0..3)
  int tj0 = (wave & 1) * 2;        // B-tile cols tj0, tj0+1

  const float* arow = As + (ti * 16 + lr) * 68 + 2 * half;
  const float* brow0 = Bs + (tj0 * 16 + lr) * 68 + 2 * half;
  const float* brow1 = Bs + ((tj0 + 1) * 16 + lr) * 68 + 2 * half;

  float4* As4 = (float4*)As;
  float4* Bs4 = (float4*)Bs;
  const float* Abase = A + ((size_t)b * N + (size_t)bi * 64) * C;
  const float* Bbase = B + ((size_t)b * N + (size_t)bj * 64) * C;

  v8f acc0 = {}, acc1 = {};
  for (int kc = 0; kc < C; kc += 64) {
    __syncthreads();               // protect previous chunk's LDS reads
    #pragma unroll
    for (int rpt = 0; rpt < 4; ++rpt) {
      int idx = t + rpt * 256;     // 1024 float4 per array per chunk
      int row = idx >> 4, c4 = idx & 15;
      As4[row * 17 + c4] = ((const float4*)(Abase + (size_t)row * C + kc))[c4];
      Bs4[row * 17 + c4] = ((const float4*)(Bbase + (size_t)row * C + kc))[c4];
    }
    __syncthreads();
    #pragma unroll 4
    for (int kk = 0; kk < 64; kk += 4) {
      v2f af, bf0, bf1;
      float2 a2 = *(const float2*)&arow[kk];
      float2 b2 = *(const float2*)&brow0[kk];
      float2 c2 = *(const float2*)&brow1[kk];
      af.x = a2.x;  af.y = a2.y;
      bf0.x = b2.x; bf0.y = b2.y;
      bf1.x = c2.x; bf1.y = c2.y;
      acc0 = __builtin_amdgcn_wmma_f32_16x16x4_f32(false, af, false, bf0,
                                                   (short)0, acc0, false, false);
      acc1 = __builtin_amdgcn_wmma_f32_16x16x4_f32(false, af, false, bf1,
                                                   (short)0, acc1, false, false);
    }
  }

  float* crow = sim + (size_t)b * N * N;
  int j0 = bj * 64 + tj0 * 16 + lr;
  int j1 = j0 + 16;
  #pragma unroll
  for (int r = 0; r < 8; ++r) {
    int i = bi * 64 + ti * 16 + r + 8 * half;
    crow[(size_t)i * N + j0] = acc0[r];
    crow[(size_t)i * N + j1] = acc1[r];
  }
}

// ---------------- Sinkhorn ----------------
__global__ __launch_bounds__(256) void k_init_uv(float* __restrict__ u,
                                                 float* __restrict__ v) {
  int id = blockIdx.x * 256 + threadIdx.x;
  if (id < BATCH * (N + 1)) { u[id] = 0.f; v[id] = 0.f; }
}

// u[i] = norm - LSE_j( {sim[i,j]/T + v[j]} U {ds + v[N]} ), one block per row
__global__ __launch_bounds__(256) void k_u_rows(const float* __restrict__ sim,
                                                const float* __restrict__ v,
                                                float* __restrict__ u,
                                                const float* __restrict__ dsp,
                                                const float* __restrict__ tempp) {
  int b = blockIdx.y, i = blockIdx.x, t = threadIdx.x;
  const float* srow = sim + ((size_t)b * N + i) * N;
  const float* vb = v + (size_t)b * (N + 1);
  float invT = 1.f / tempp[0];
  float m = -INFINITY, s = 0.f;
  for (int j = t; j < N; j += 256) {
    float x = srow[j] * invT + vb[j];
    if (x > m) { s = s * expf(m - x) + 1.f; m = x; }
    else       { s += expf(x - m); }
  }
  __shared__ float mm[256], ss[256];
  mm[t] = m; ss[t] = s;
  __syncthreads();
  for (int o = 128; o > 0; o >>= 1) {
    if (t < o) {
      float m2 = mm[t + o], s2 = ss[t + o];
      float M = fmaxf(mm[t], m2);
      ss[t] = (M == -INFINITY) ? 0.f : (ss[t] * expf(mm[t] - M) + s2 * expf(m2 - M));
      mm[t] = M;
    }
    __syncthreads();
  }
  if (t == 0) {
    float x = dsp[0] + vb[N];                       // dustbin column term
    float M = fmaxf(mm[0], x);
    float S = ss[0] * expf(mm[0] - M) + expf(x - M);
    float norm = -logf((float)(2 * N));
    u[(size_t)b * (N + 1) + i] = norm - (M + logf(S));
  }
}

// u[N] = log(N)+norm - LSE( {ds + v[j]} U {v[N]} ), one block per batch
__global__ __launch_bounds__(256) void k_u_dust(const float* __restrict__ v,
                                                float* __restrict__ u,
                                                const float* __restrict__ dsp) {
  int b = blockIdx.x, t = threadIdx.x;
  const float* vb = v + (size_t)b * (N + 1);
  float m = -INFINITY, s = 0.f;
  for (int j = t; j < N; j += 256) {
    float x = vb[j];
    if (x > m) { s = s * expf(m - x) + 1.f; m = x; }
    else       { s += expf(x - m); }
  }
  __shared__ float mm[256], ss[256];
  mm[t] = m; ss[t] = s;
  __syncthreads();
  for (int o = 128; o > 0; o >>= 1) {
    if (t < o) {
      float m2 = mm[t + o], s2 = ss[t + o];
      float M = fmaxf(mm[t], m2);
      ss[t] = (M == -INFINITY) ? 0.f : (ss[t] * expf(mm[t] - M) + s2 * expf(m2 - M));
      mm[t] = M;
    }
    __syncthreads();
  }
  if (t == 0) {
    float mA = mm[0] + dsp[0];              // shift set {v[j]} by ds
    float x = vb[N];                        // + Z[M][N]=0 entry
    float M = fmaxf(mA, x);
    float S = ss[0] * expf(mA - M) + expf(x - M);
    float norm = -logf((float)(2 * N));
    u[(size_t)b * (N + 1) + N] = (logf((float)N) + norm) - (M + logf(S));
  }
}

// Column LSE partials: grid (N/256, RCHUNK, B); each thread handles one column
// over a 256-row chunk (coalesced across lanes), emitting an online (m,s) pair.
__global__ __launch_bounds__(256) void k_v_cols_part(const float* __restrict__ sim,
                                                     const float* __restrict__ u,
                                                     float* __restrict__ pm,
                                                     float* __restrict__ ps,
                                                     const float* __restrict__ tempp) {
  int b = blockIdx.z, rc = blockIdx.y;
  int j = blockIdx.x * 256 + threadIdx.x;
  const float* sb = sim + (size_t)b * N * N;
  const float* ub = u + (size_t)b * (N + 1);
  float invT = 1.f / tempp[0];
  float m = -INFINITY, s = 0.f;
  int i0 = rc * (N / RCHUNK), i1 = i0 + (N / RCHUNK);
  for (int i = i0; i < i1; ++i) {
    float x = sb[(size_t)i * N + j] * invT + ub[i];
    if (x > m) { s = s * expf(m - x) + 1.f; m = x; }
    else       { s += expf(x - m); }
  }
  size_t o = ((size_t)b * RCHUNK + rc) * N + j;
  pm[o] = m;
  ps[o] = s;
}

// Combine partials + dustbin row: v[j] = norm - LSE
__global__ __launch_bounds__(256) void k_v_cols_comb(const float* __restrict__ pm,
                                                     const float* __restrict__ ps,
                                                     const float* __restrict__ u,
                                                     float* __restrict__ v,
                                                     const float* __restrict__ dsp) {
  int b = blockIdx.y;
  int j = blockIdx.x * 256 + threadIdx.x;
  float m = -INFINITY, s = 0.f;
  #pragma unroll
  for (int rc = 0; rc < RCHUNK; ++rc) {
    size_t o = ((size_t)b * RCHUNK + rc) * N + j;
    float m2 = pm[o], s2 = ps[o];
    float M = fmaxf(m, m2);
    s = (M == -INFINITY) ? 0.f : (s * expf(m - M) + s2 * expf(m2 - M));
    m = M;
  }
  float x = dsp[0] + u[(size_t)b * (N + 1) + N];   // dustbin row term
  if (x > m) { s = s * expf(m - x) + 1.f; m = x; }
  else       { s += expf(x - m); }
  float norm = -logf((float)(2 * N));
  v[(size_t)b * (N + 1) + j] = norm - (m + logf(s));
}

// v[N] = log(M)+norm - LSE( {ds + u[i]} U {u[N]} )
__global__ __launch_bounds__(256) void k_v_dust(const float* __restrict__ u,
                                                float* __restrict__ v,
                                                const float* __restrict__ dsp) {
  int b = blockIdx.x, t = threadIdx.x;
  const float* ub = u + (size_t)b * (N + 1);
  float m = -INFINITY, s = 0.f;
  for (int i = t; i < N; i += 256) {
    float x = ub[i];
    if (x > m) { s = s * expf(m - x) + 1.f; m = x; }
    else       { s += expf(x - m); }
  }
  __shared__ float mm[256], ss[256];
  mm[t] = m; ss[t] = s;
  __syncthreads();
  for (int o = 128; o > 0; o >>= 1) {
    if (t < o) {
      float m2 = mm[t + o], s2 = ss[t + o];
      float M = fmaxf(mm[t], m2);
      ss[t] = (M == -INFINITY) ? 0.f : (ss[t] * expf(mm[t] - M) + s2 * expf(m2 - M));
      mm[t] = M;
    }
    __syncthreads();
  }
  if (t == 0) {
    float mA = mm[0] + dsp[0];
    float x = ub[N];
    float M = fmaxf(mA, x);
    float S = ss[0] * expf(mA - M) + expf(x - M);
    float norm = -logf((float)(2 * N));
    v[(size_t)b * (N + 1) + N] = (logf((float)N) + norm) - (M + logf(S));
  }
}

// ---------------- finalize: ot_prob row -> entropy + top-8 + gather pos ----------------
__global__ __launch_bounds__(256) void k_finalize(const float* __restrict__ sim,
                                                  const float* __restrict__ u,
                                                  const float* __restrict__ v,
                                                  const float* __restrict__ posB,
                                                  const float* __restrict__ tempp,
                                                  float* __restrict__ ent_out,
                                                  float* __restrict__ tv_out,
                                                  int* __restrict__ ti_out,
                                                  float* __restrict__ tp_out) {
  __shared__ float q[N];
  __shared__ float red[256];
  __shared__ int   redi[256];
  __shared__ float Ssh;
  int b = blockIdx.y, i = blockIdx.x, t = threadIdx.x;
  const float* srow = sim + ((size_t)b * N + i) * N;
  const float* vb = v + (size_t)b * (N + 1);
  float ui = u[(size_t)b * (N + 1) + i];
  float invT = 1.f / tempp[0];

  float sl = 0.f;
  for (int j = t; j < N; j += 256) {
    float p = expf(srow[j] * invT + ui + vb[j]);
    q[j] = p;
    sl += p;
  }
  red[t] = sl;
  __syncthreads();
  for (int o = 128; o > 0; o >>= 1) {
    if (t < o) red[t] += red[t + o];
    __syncthreads();
  }
  if (t == 0) Ssh = red[0];
  __syncthreads();
  float invS = 1.f / (Ssh + 1e-8f);

  float el = 0.f;
  for (int j = t; j < N; j += 256) {
    float p = q[j] * invS;
    el += p * logf(p + 1e-8f);
  }
  red[t] = el;
  __syncthreads();
  for (int o = 128; o > 0; o >>= 1) {
    if (t < o) red[t] += red[t + o];
    __syncthreads();
  }
  if (t == 0) ent_out[(size_t)b * N + i] = -red[0];
  __syncthreads();

  // iterative top-8 with lowest-index tie-break (lax.top_k compatible)
  for (int k = 0; k < TOPK; ++k) {
    float bm = -INFINITY; int bi = N;
    for (int j = t; j < N; j += 256) {
      float val = q[j];
      if (val > bm) { bm = val; bi = j; }
    }
    red[t] = bm; redi[t] = bi;
    __syncthreads();
    for (int o = 128; o > 0; o >>= 1) {
      if (t < o) {
        if (red[t + o] > red[t] ||
            (red[t + o] == red[t] && redi[t + o] < redi[t])) {
          red[t] = red[t + o]; redi[t] = redi[t + o];
        }
      }
      __syncthreads();
    }
    if (t == 0) {
      int idx = redi[0];
      size_t base = ((size_t)b * N + i) * TOPK + k;
      tv_out[base] = red[0];
      ti_out[base] = idx;
      tp_out[base * 2 + 0] = posB[((size_t)b * N + idx) * 2 + 0];
      tp_out[base * 2 + 1] = posB[((size_t)b * N + idx) * 2 + 1];
      q[idx] = -INFINITY;
    }
    __syncthreads();
  }
}

// ---------------- geometric validator (7x7 window, count_include_pad=False) ----------
__global__ __launch_bounds__(256) void k_geo(const float* __restrict__ posA,
                                             const float* __restrict__ tp,
                                             float* __restrict__ geo_out) {
  int gid = blockIdx.x * 256 + threadIdx.x;   // B*N*K threads
  int k = gid & (TOPK - 1);
  int n = (gid >> 3) & (N - 1);
  int b = gid >> 15;
  int y = n >> 6, x = n & (WDIM - 1);
  float s0 = 0.f, s1 = 0.f, q0 = 0.f, q1 = 0.f;
  int cnt = 0;
  for (int dy = -RADIUS; dy <= RADIUS; ++dy) {
    int ny = y + dy;
    if (ny < 0 || ny >= HDIM) continue;
    for (int dx = -RADIUS; dx <= RADIUS; ++dx) {
      int nx = x + dx;
      if (nx < 0 || nx >= WDIM) continue;
      int nn = ny * WDIM + nx;
      size_t pb = ((size_t)b * N + nn) * TOPK + k;
      float d0 = tp[pb * 2 + 0] - posA[((size_t)b * N + nn) * 2 + 0];
      float d1 = tp[pb * 2 + 1] - posA[((size_t)b * N + nn) * 2 + 1];
      s0 += d0; s1 += d1; q0 += d0 * d0; q1 += d1 * d1;
      ++cnt;
    }
  }
  float ic = 1.f / (float)cnt;
  float m0 = s0 * ic, m1 = s1 * ic;
  float v0 = fmaxf(q0 * ic - m0 * m0, 0.f);
  float v1 = fmaxf(q1 * ic - m1 * m1, 0.f);
  geo_out[gid] = 1.f / (1.f + (v0 + v1) * 100.f);
}

// ---------------- softmax refine -> warp positions ----------------
__global__ __launch_bounds__(256) void k_refine(const float* __restrict__ tv,
                                                const float* __restrict__ tp,
                                                const float* __restrict__ geo,
                                                const float* __restrict__ gwp,
                                                const float* __restrict__ tempp,
                                                float* __restrict__ warp_out) {
  int id = blockIdx.x * 256 + threadIdx.x;    // B*N threads
  if (id >= BATCH * N) return;
  float gw = fminf(fmaxf(gwp[0], 0.f), 2.f);
  float invT = 1.f / tempp[0];
  size_t base = (size_t)id * TOPK;
  float comb[TOPK];
  float mx = -INFINITY;
  #pragma unroll
  for (int k = 0; k < TOPK; ++k) {
    comb[k] = (tv[base + k] + gw * geo[base + k]) * invT;
    mx = fmaxf(mx, comb[k]);
  }
  float se = 0.f;
  #pragma unroll
  for (int k = 0; k < TOPK; ++k) { comb[k] = expf(comb[k] - mx); se += comb[k]; }
  float inv = 1.f / se;
  float wx = 0.f, wy = 0.f;
  #pragma unroll
  for (int k = 0; k < TOPK; ++k) {
    float w = comb[k] * inv;
    wx += w * tp[(base + k) * 2 + 0];
    wy += w * tp[(base + k) * 2 + 1];
  }
  warp_out[(size_t)id * 2 + 0] = wx;
  warp_out[(size_t)id * 2 + 1] = wy;
}

extern "C" void kernel_launch(void* const* d_in, const int* in_sizes, int n_in,
                              void* d_out, int out_size, void* d_ws, size_t ws_size,
                              hipStream_t stream) {
  const float* featA = (const float*)d_in[0];
  const float* featB = (const float*)d_in[1];
  const float* posA  = (const float*)d_in[2];
  const float* posB  = (const float*)d_in[3];
  const float* dsp   = (const float*)d_in[4];
  const float* gwp   = (const float*)d_in[5];
  const float* tmp   = (const float*)d_in[6];
  (void)in_sizes; (void)n_in; (void)out_size; (void)ws_size;

  float* out = (float*)d_out;
  float* ws  = (float*)d_ws;

  float* normA = ws + W_NA;
  float* normB = ws + W_NB;
  float* u     = ws + W_U;
  float* v     = ws + W_V;
  float* tv    = ws + W_TV;
  int*   ti    = (int*)(ws + W_TI);
  float* tp    = ws + W_TP;
  float* pm    = ws + W_PM;
  float* ps    = ws + W_PS;

  float* sim   = out + O_SIM;
  float* ent   = out + O_ENT;
  float* geo   = out + O_GEO;
  float* warp  = out + O_WARP;

  // 1) normalize features
  k_rownorm<<<BATCH * N, 128, 0, stream>>>(featA, normA);
  k_rownorm<<<BATCH * N, 128, 0, stream>>>(featB, normB);

  // 2) raw_sim via fp32 WMMA, LDS-tiled 64x64 blocks (writes d_out slice; doubles as Z)
  k_gemm_wmma<<<BATCH * (N / 64) * (N / 64), 256, 0, stream>>>(normA, normB, sim);

  // 3) Sinkhorn
  k_init_uv<<<(BATCH * (N + 1) + 255) / 256, 256, 0, stream>>>(u, v);
  for (int it = 0; it < SINK_ITERS; ++it) {
    k_u_rows<<<dim3(N, BATCH), 256, 0, stream>>>(sim, v, u, dsp, tmp);
    k_u_dust<<<BATCH, 256, 0, stream>>>(v, u, dsp);
    k_v_cols_part<<<dim3(N / 256, RCHUNK, BATCH), 256, 0, stream>>>(sim, u, pm, ps, tmp);
    k_v_cols_comb<<<dim3(N / 256, BATCH), 256, 0, stream>>>(pm, ps, u, v, dsp);
    k_v_dust<<<BATCH, 256, 0, stream>>>(u, v, dsp);
  }

  // 4) per-row ot_prob -> entropy, top-8, gathered candidate positions
  k_finalize<<<dim3(N, BATCH), 256, 0, stream>>>(sim, u, v, posB, tmp, ent, tv, ti, tp);

  // 5) geometric validation (writes geo_scores output)
  k_geo<<<(BATCH * N * TOPK) / 256, 256, 0, stream>>>(posA, tp, geo);

  // 6) softmax refine (writes refined_warp output)
  k_refine<<<(BATCH * N + 255) / 256, 256, 0, stream>>>(tv, tp, geo, gwp, tmp, warp);
}